// NonRigidICP_2173253452333
// MI455X (gfx1250) — compile-verified
//
#include <hip/hip_runtime.h>
#include <hip/hip_bf16.h>
#include <math.h>

#define M_NODES 128
#define N_CORR  16384
#define K_ANCH  4
#define N_EDGES 1024
#define NP      768           // M_NODES * 6 parameters
#define LM_LAMBDA 0.01f

typedef __attribute__((ext_vector_type(2))) float v2f;
typedef __attribute__((ext_vector_type(8))) float v8f;

// skew(v) = [[0,-z,y],[z,0,-x],[-y,x,0]]
__device__ __forceinline__ float skew_elem(const float v[3], int i, int j) {
    if (i == 0) return (j == 1) ? -v[2] : ((j == 2) ?  v[1] : 0.0f);
    if (i == 1) return (j == 0) ?  v[2] : ((j == 2) ? -v[0] : 0.0f);
    return              (j == 0) ? -v[1] : ((j == 1) ?  v[0] : 0.0f);
}

// ---------------------------------------------------------------------------
// State / system init
// ---------------------------------------------------------------------------
__global__ void init_state_kernel(float* __restrict__ R, float* __restrict__ T) {
    int m = threadIdx.x;
    if (m < M_NODES) {
        for (int j = 0; j < 9; ++j) R[m * 9 + j] = (j == 0 || j == 4 || j == 8) ? 1.0f : 0.0f;
        for (int j = 0; j < 3; ++j) T[m * 3 + j] = 0.0f;
    }
}

__global__ void init_A_kernel(float* __restrict__ A, float* __restrict__ bvec) {
    int i = blockIdx.x * blockDim.x + threadIdx.x;
    if (i < NP * NP) {
        int r = i / NP, c = i % NP;
        A[i] = (r == c) ? LM_LAMBDA : 0.0f;
    }
    if (i < NP) bvec[i] = 0.0f;
}

// ---------------------------------------------------------------------------
// Correspondence Gram accumulation: one workgroup per anchor base (128 WGs).
// Each wave streams its corrs; per corr a 3x24 local Jacobian (padded 4x32)
// feeds V_WMMA_F32_16X16X4_F32 chains building the 32x32 local Gram.
// ---------------------------------------------------------------------------
__global__ __launch_bounds__(256) void corr_gram_kernel(
    const float* __restrict__ src_nodes, const float* __restrict__ src_corr,
    const float* __restrict__ tgt_corr,  const int*   __restrict__ anch_idx,
    const float* __restrict__ anch_w,    const float* __restrict__ corr_w,
    const float* __restrict__ R,         const float* __restrict__ T,
    float* __restrict__ A,               float* __restrict__ bvec)
{
    __shared__ float sR[4][9], sT[4][3], sN[4][3];
    __shared__ float sA[32][32];
    __shared__ float sB[32];

    const int b   = blockIdx.x;          // bucket = base node
    const int tid = threadIdx.x;

    if (tid < 36)               { int a = tid / 9, e = tid % 9; int n = (b + a) & (M_NODES - 1); sR[a][e] = R[n * 9 + e]; }
    else if (tid < 48)          { int q = tid - 36; int a = q / 3, e = q % 3; int n = (b + a) & (M_NODES - 1); sT[a][e] = T[n * 3 + e]; }
    else if (tid < 60)          { int q = tid - 48; int a = q / 3, e = q % 3; int n = (b + a) & (M_NODES - 1); sN[a][e] = src_nodes[n * 3 + e]; }
    for (int i = tid; i < 32 * 32; i += blockDim.x) ((float*)sA)[i] = 0.0f;
    if (tid < 32) sB[tid] = 0.0f;
    __syncthreads();

    const int lane   = tid & 31;
    const int wave   = tid >> 5;
    const int nwaves = blockDim.x >> 5;
    const int m16    = lane & 15;
    const int kb     = (lane < 16) ? 0 : 2;   // lane-half selects K-pair {0,1} vs {2,3}

    v8f acc00 = {}, acc01 = {}, acc10 = {}, acc11 = {};

    for (int c = wave; c < N_CORR; c += nwaves) {
        int base = anch_idx[c * K_ANCH];          // uniform across the wave
        if (base != b) continue;

        float p0 = src_corr[c * 3 + 0], p1 = src_corr[c * 3 + 1], p2 = src_corr[c * 3 + 2];
        float r0 = -tgt_corr[c * 3 + 0], r1 = -tgt_corr[c * 3 + 1], r2 = -tgt_corr[c * 3 + 2];
        float cw = sqrtf(corr_w[c]);

        float wp[4], Rd[4][3];
        for (int a = 0; a < 4; ++a) {
            float aw = anch_w[c * 4 + a];
            wp[a] = cw * aw;
            float l0 = p0 - sN[a][0], l1 = p1 - sN[a][1], l2 = p2 - sN[a][2];
            Rd[a][0] = sR[a][0] * l0 + sR[a][1] * l1 + sR[a][2] * l2;
            Rd[a][1] = sR[a][3] * l0 + sR[a][4] * l1 + sR[a][5] * l2;
            Rd[a][2] = sR[a][6] * l0 + sR[a][7] * l1 + sR[a][8] * l2;
            r0 += aw * (Rd[a][0] + sN[a][0] + sT[a][0]);
            r1 += aw * (Rd[a][1] + sN[a][1] + sT[a][1]);
            r2 += aw * (Rd[a][2] + sN[a][2] + sT[a][2]);
        }
        float r[3] = { r0, r1, r2 };

        // J[k][col], col layout: [rot a0..a3 (12) | trans a0..a3 (12) | zero pad (8)]
        auto jval = [&](int k, int col) -> float {
            if (k >= 3 || col >= 24) return 0.0f;
            if (col < 12) { int a = col / 3, j = col % 3; return -wp[a] * skew_elem(Rd[a], k, j); }
            int tc = col - 12; int a = tc / 3, j = tc % 3;
            return (k == j) ? wp[a] : 0.0f;
        };

        v2f f0, f1;   // identical register image serves as both A (J^T tile) and B (J tile)
        f0.x = jval(kb,     m16);      f0.y = jval(kb + 1,     m16);
        f1.x = jval(kb, 16 + m16);     f1.y = jval(kb + 1, 16 + m16);

        acc00 = __builtin_amdgcn_wmma_f32_16x16x4_f32(false, f0, false, f0, (short)0, acc00, false, false);
        acc01 = __builtin_amdgcn_wmma_f32_16x16x4_f32(false, f0, false, f1, (short)0, acc01, false, false);
        acc10 = __builtin_amdgcn_wmma_f32_16x16x4_f32(false, f1, false, f0, (short)0, acc10, false, false);
        acc11 = __builtin_amdgcn_wmma_f32_16x16x4_f32(false, f1, false, f1, (short)0, acc11, false, false);

        // b -= J^T r  (each lane-half contributes its K-pair's partial dot)
        float rk0 = r[kb];
        float rk1 = (kb + 1 < 3) ? r[kb + 1] : 0.0f;
        atomicAdd(&sB[m16], -(f0.x * rk0 + f0.y * rk1));
        if (m16 < 8) atomicAdd(&sB[16 + m16], -(f1.x * rk0 + f1.y * rk1));
    }

    // Flush wave accumulators to the LDS Gram. D(m,n): lane = n + 16*(m>=8), vgpr = m&7
    {
        int nloc = m16;
        int mh   = (lane >> 4) * 8;
        for (int vr = 0; vr < 8; ++vr) {
            atomicAdd(&sA[     mh + vr][     nloc], acc00[vr]);
            atomicAdd(&sA[     mh + vr][16 + nloc], acc01[vr]);
            atomicAdd(&sA[16 + mh + vr][     nloc], acc10[vr]);
            atomicAdd(&sA[16 + mh + vr][16 + nloc], acc11[vr]);
        }
    }
    __syncthreads();

    // Scatter 24x24 local Gram + 24-vector into the global system.
    auto gmap = [&](int l) -> int {
        if (l < 12) { int a = l / 3, j = l % 3; int n = (b + a) & (M_NODES - 1); return 3 * n + j; }
        int tc = l - 12; int a = tc / 3, j = tc % 3; int n = (b + a) & (M_NODES - 1);
        return 384 + 3 * n + j;
    };
    for (int idx = tid; idx < 24 * 24; idx += blockDim.x) {
        int lr = idx / 24, lc = idx % 24;
        atomicAdd(&A[gmap(lr) * NP + gmap(lc)], sA[lr][lc]);
    }
    if (tid < 24) atomicAdd(&bvec[gmap(tid)], sB[tid]);
}

// ---------------------------------------------------------------------------
// ARAP edge Gram accumulation (E=1024, one thread per edge, atomics).
// ---------------------------------------------------------------------------
__global__ void arap_gram_kernel(
    const float* __restrict__ nodes, const int* __restrict__ edges,
    const float* __restrict__ ew,    const float* __restrict__ R,
    const float* __restrict__ T,     float* __restrict__ A, float* __restrict__ bvec)
{
    int e = blockIdx.x * blockDim.x + threadIdx.x;
    if (e >= N_EDGES) return;
    int u = edges[e * 2 + 0], v = edges[e * 2 + 1];
    float w = sqrtf(ew[e]);

    float d0 = nodes[v * 3 + 0] - nodes[u * 3 + 0];
    float d1 = nodes[v * 3 + 1] - nodes[u * 3 + 1];
    float d2 = nodes[v * 3 + 2] - nodes[u * 3 + 2];
    float Rd[3];
    Rd[0] = R[u * 9 + 0] * d0 + R[u * 9 + 1] * d1 + R[u * 9 + 2] * d2;
    Rd[1] = R[u * 9 + 3] * d0 + R[u * 9 + 4] * d1 + R[u * 9 + 5] * d2;
    Rd[2] = R[u * 9 + 6] * d0 + R[u * 9 + 7] * d1 + R[u * 9 + 8] * d2;

    float rd[3][3];
    for (int i = 0; i < 3; ++i)
        for (int j = 0; j < 3; ++j)
            rd[i][j] = -w * skew_elem(Rd, i, j);

    float r[3];
    for (int i = 0; i < 3; ++i)
        r[i] = w * (Rd[i] + T[u * 3 + i] + nodes[u * 3 + i] - nodes[v * 3 + i] - T[v * 3 + i]);

    int cu = 3 * u, tu = 384 + 3 * u, tv = 384 + 3 * v;
    for (int i = 0; i < 3; ++i) {
        for (int j = 0; j < 3; ++j) {
            float s = rd[0][i] * rd[0][j] + rd[1][i] * rd[1][j] + rd[2][i] * rd[2][j];
            atomicAdd(&A[(cu + i) * NP + cu + j], s);
            float x = w * rd[j][i];
            atomicAdd(&A[(cu + i) * NP + tu + j],  x);
            atomicAdd(&A[(tu + j) * NP + cu + i],  x);
            atomicAdd(&A[(cu + i) * NP + tv + j], -x);
            atomicAdd(&A[(tv + j) * NP + cu + i], -x);
        }
    }
    float w2 = w * w;
    for (int i = 0; i < 3; ++i) {
        atomicAdd(&A[(tu + i) * NP + tu + i],  w2);
        atomicAdd(&A[(tv + i) * NP + tv + i],  w2);
        atomicAdd(&A[(tu + i) * NP + tv + i], -w2);
        atomicAdd(&A[(tv + i) * NP + tu + i], -w2);
    }
    for (int i = 0; i < 3; ++i) {
        float s = rd[0][i] * r[0] + rd[1][i] * r[1] + rd[2][i] * r[2];
        atomicAdd(&bvec[cu + i], -s);
        atomicAdd(&bvec[tu + i], -w * r[i]);
        atomicAdd(&bvec[tv + i],  w * r[i]);
    }
}

// ---------------------------------------------------------------------------
// Single-workgroup Cholesky (A SPD) + forward/backward substitution, in-place.
// ---------------------------------------------------------------------------
__global__ __launch_bounds__(1024) void solve_kernel(float* __restrict__ A, float* __restrict__ bvec) {
    const int n = NP, tid = threadIdx.x, TH = blockDim.x;
    for (int k = 0; k < n; ++k) {
        if (tid == 0) A[k * n + k] = sqrtf(A[k * n + k]);
        __syncthreads();
        float d = A[k * n + k];
        for (int i = k + 1 + tid; i < n; i += TH) A[i * n + k] /= d;
        __syncthreads();
        for (int i = k + 1 + tid; i < n; i += TH) {
            float lik = A[i * n + k];
            for (int j = k + 1; j <= i; ++j) A[i * n + j] -= lik * A[j * n + k];
        }
        __syncthreads();
    }
    for (int k = 0; k < n; ++k) {            // L y = b
        if (tid == 0) bvec[k] /= A[k * n + k];
        __syncthreads();
        float yk = bvec[k];
        for (int i = k + 1 + tid; i < n; i += TH) bvec[i] -= A[i * n + k] * yk;
        __syncthreads();
    }
    for (int k = n - 1; k >= 0; --k) {       // L^T x = y
        if (tid == 0) bvec[k] /= A[k * n + k];
        __syncthreads();
        float xk = bvec[k];
        for (int i = tid; i < k; i += TH) bvec[i] -= A[k * n + i] * xk;
        __syncthreads();
    }
}

// ---------------------------------------------------------------------------
// Exponential-map update of node transforms.
// ---------------------------------------------------------------------------
__global__ void update_kernel(const float* __restrict__ x, float* __restrict__ R, float* __restrict__ T) {
    int m = blockIdx.x * blockDim.x + threadIdx.x;
    if (m >= M_NODES) return;
    float w0 = x[3 * m + 0], w1 = x[3 * m + 1], w2 = x[3 * m + 2];
    float th2 = w0 * w0 + w1 * w1 + w2 * w2;
    float Ac, Bc;
    if (th2 < 1e-12f) { Ac = 1.0f; Bc = 0.5f; }
    else { float th = sqrtf(th2); Ac = sinf(th) / th; Bc = (1.0f - cosf(th)) / th2; }
    float K[9] = { 0.0f, -w2, w1,  w2, 0.0f, -w0,  -w1, w0, 0.0f };
    float K2[9];
    for (int i = 0; i < 3; ++i)
        for (int j = 0; j < 3; ++j) {
            float s = 0.0f;
            for (int kk = 0; kk < 3; ++kk) s += K[i * 3 + kk] * K[kk * 3 + j];
            K2[i * 3 + j] = s;
        }
    float Ri[9];
    for (int i = 0; i < 3; ++i)
        for (int j = 0; j < 3; ++j)
            Ri[i * 3 + j] = ((i == j) ? 1.0f : 0.0f) + Ac * K[i * 3 + j] + Bc * K2[i * 3 + j];
    float Ro[9];
    for (int j = 0; j < 9; ++j) Ro[j] = R[m * 9 + j];
    for (int i = 0; i < 3; ++i)
        for (int j = 0; j < 3; ++j) {
            float s = 0.0f;
            for (int kk = 0; kk < 3; ++kk) s += Ri[i * 3 + kk] * Ro[kk * 3 + j];
            R[m * 9 + i * 3 + j] = s;
        }
    for (int i = 0; i < 3; ++i) T[m * 3 + i] += x[384 + 3 * m + i];
}

__global__ void writeout_kernel(const float* __restrict__ R, const float* __restrict__ T, float* __restrict__ out) {
    int m = blockIdx.x * blockDim.x + threadIdx.x;
    if (m >= M_NODES) return;
    float* o = out + m * 16;
    for (int i = 0; i < 3; ++i) {
        for (int j = 0; j < 3; ++j) o[i * 4 + j] = R[m * 9 + i * 3 + j];
        o[i * 4 + 3] = T[m * 3 + i];
    }
    o[12] = 0.0f; o[13] = 0.0f; o[14] = 0.0f; o[15] = 1.0f;
}

// ---------------------------------------------------------------------------
extern "C" void kernel_launch(void* const* d_in, const int* in_sizes, int n_in,
                              void* d_out, int out_size, void* d_ws, size_t ws_size,
                              hipStream_t stream) {
    const float* src_nodes = (const float*)d_in[0];
    const float* src_corr  = (const float*)d_in[1];
    const float* tgt_corr  = (const float*)d_in[2];
    const int*   anch_idx  = (const int*)  d_in[3];
    const float* anch_w    = (const float*)d_in[4];
    const int*   edges     = (const int*)  d_in[5];
    const float* corr_w    = (const float*)d_in[6];
    const float* edge_w    = (const float*)d_in[7];

    float* ws = (float*)d_ws;
    float* R  = ws;                    // 128*9
    float* T  = R + M_NODES * 9;       // 128*3
    float* A  = T + M_NODES * 3;       // 768*768
    float* bv = A + NP * NP;           // 768
    float* out = (float*)d_out;

    init_state_kernel<<<1, 128, 0, stream>>>(R, T);
    for (int it = 0; it < 2; ++it) {
        init_A_kernel<<<(NP * NP + 255) / 256, 256, 0, stream>>>(A, bv);
        corr_gram_kernel<<<M_NODES, 256, 0, stream>>>(src_nodes, src_corr, tgt_corr,
                                                      anch_idx, anch_w, corr_w, R, T, A, bv);
        arap_gram_kernel<<<(N_EDGES + 255) / 256, 256, 0, stream>>>(src_nodes, edges, edge_w, R, T, A, bv);
        solve_kernel<<<1, 1024, 0, stream>>>(A, bv);
        update_kernel<<<1, 128, 0, stream>>>(bv, R, T);
    }
    writeout_kernel<<<1, 128, 0, stream>>>(R, T, out);
}